// GraphSAGEClassifier_13597866459807
// MI455X (gfx1250) — compile-verified
//
#include <hip/hip_runtime.h>
#include <hip/hip_bf16.h>

typedef __attribute__((ext_vector_type(2))) float v2f;
typedef __attribute__((ext_vector_type(8))) float v8f;

#define DFEAT 128
#define LDSPITCH 132   // 132 % 64 banks = 4 -> 16/32 rows hit distinct banks on column reads
#define MROWS 32       // nodes per block in the fused layer kernel

// ---------------------------------------------------------------- utilities
__global__ __launch_bounds__(256) void zero_kernel(float* __restrict__ p, long long n) {
    long long i = (long long)blockIdx.x * blockDim.x + threadIdx.x;
    long long stride = (long long)gridDim.x * blockDim.x;
    for (; i < n; i += stride) p[i] = 0.0f;
}

__global__ __launch_bounds__(256) void degree_kernel(const int* __restrict__ dst, int E,
                                                     float* __restrict__ cnt) {
    int e = blockIdx.x * blockDim.x + threadIdx.x;
    if (e < E) atomicAdd(cnt + dst[e], 1.0f);
}

// One wave per edge: lane L moves bytes [L*16, L*16+16) of the 512-byte row.
// Whole working set (~150 MB) is L2-resident on MI455X (192 MB L2).
__global__ __launch_bounds__(256) void scatter_kernel(const float* __restrict__ x,
                                                      const int* __restrict__ src,
                                                      const int* __restrict__ dst,
                                                      float* agg, int E) {
    int t = blockIdx.x * blockDim.x + threadIdx.x;
    int e = t >> 5;
    if (e >= E) return;
    int lane = t & 31;
    int s = src[e];
    int d = dst[e];
    const float4 v = ((const float4*)(x + (size_t)s * DFEAT))[lane];
    float* p = agg + (size_t)d * DFEAT + lane * 4;
    atomicAdd(p + 0, v.x);
    atomicAdd(p + 1, v.y);
    atomicAdd(p + 2, v.z);
    atomicAdd(p + 3, v.w);
}

// ------------------------------------------------- fused mean + dual GEMM + ReLU
// Block = 32 nodes x 128 out-features; 8 waves. Each wave owns one 16-col tile
// and TWO 16x16 accumulators (rows 0-15 / 16-31), so every B-fragment load
// feeds two back-to-back v_wmma_f32_16x16x4_f32 ops.
// out = ReLU( (agg/max(cnt,1)) @ Wl^T + bl + xin @ Wr^T ).  out may alias agg.
__global__ __launch_bounds__(256) void sage_layer_kernel(
    const float* __restrict__ xin, const float* agg, const float* __restrict__ cnt,
    const float* __restrict__ Wl, const float* __restrict__ bl,
    const float* __restrict__ Wr, float* out, int n) {
    __shared__ float xs[MROWS][LDSPITCH];
    __shared__ float ms[MROWS][LDSPITCH];
    __shared__ float inv[MROWS];

    const int base = blockIdx.x * MROWS;
    const int tid  = threadIdx.x;

    if (tid < MROWS) {
        int r = base + tid;
        float c = (r < n) ? cnt[r] : 1.0f;
        inv[tid] = 1.0f / fmaxf(c, 1.0f);
    }
    __syncthreads();

    // stage x rows and mean rows into LDS (32*128 elements each)
    for (int i = tid; i < MROWS * DFEAT; i += 256) {
        int row = i >> 7, col = i & (DFEAT - 1);
        int r = base + row;
        int rc = (r < n) ? r : (n - 1);
        xs[row][col] = xin[(size_t)rc * DFEAT + col];
        ms[row][col] = agg[(size_t)rc * DFEAT + col] * inv[row];
    }
    __syncthreads();

    const int wave  = tid >> 5;          // 0..7  -> 16-column tile
    const int lane  = tid & 31;
    const int half  = lane >> 4;         // lane half selects K+2 / M+8
    const int arow  = lane & 15;         // A: lane = M row
    const int ncol  = wave * 16 + (lane & 15);  // B/C/D: lane = N column

    v8f c0 = {0.f, 0.f, 0.f, 0.f, 0.f, 0.f, 0.f, 0.f};
    v8f c1 = {0.f, 0.f, 0.f, 0.f, 0.f, 0.f, 0.f, 0.f};

    // mean @ Wl^T   (two M-tiles per B fragment)
    for (int k0 = 0; k0 < DFEAT; k0 += 4) {
        int k = k0 + 2 * half;
        v2f a0, a1, b;
        b.x  = Wl[(size_t)ncol * DFEAT + k];      // B[k][n] = Wl[n][k]
        b.y  = Wl[(size_t)ncol * DFEAT + k + 1];
        a0.x = ms[arow][k];
        a0.y = ms[arow][k + 1];
        a1.x = ms[16 + arow][k];
        a1.y = ms[16 + arow][k + 1];
        c0 = __builtin_amdgcn_wmma_f32_16x16x4_f32(false, a0, false, b, (short)0, c0,
                                                   false, false);
        c1 = __builtin_amdgcn_wmma_f32_16x16x4_f32(false, a1, false, b, (short)0, c1,
                                                   false, false);
    }
    // x @ Wr^T (accumulate into same tiles)
    for (int k0 = 0; k0 < DFEAT; k0 += 4) {
        int k = k0 + 2 * half;
        v2f a0, a1, b;
        b.x  = Wr[(size_t)ncol * DFEAT + k];
        b.y  = Wr[(size_t)ncol * DFEAT + k + 1];
        a0.x = xs[arow][k];
        a0.y = xs[arow][k + 1];
        a1.x = xs[16 + arow][k];
        a1.y = xs[16 + arow][k + 1];
        c0 = __builtin_amdgcn_wmma_f32_16x16x4_f32(false, a0, false, b, (short)0, c0,
                                                   false, false);
        c1 = __builtin_amdgcn_wmma_f32_16x16x4_f32(false, a1, false, b, (short)0, c1,
                                                   false, false);
    }

    const float bias = bl[ncol];
    const int m0 = half * 8;             // C/D layout: vgpr j -> M = j + 8*half
    for (int j = 0; j < 8; ++j) {
        int r0 = base + m0 + j;
        int r1 = base + 16 + m0 + j;
        if (r0 < n) out[(size_t)r0 * DFEAT + ncol] = fmaxf(c0[j] + bias, 0.0f);
        if (r1 < n) out[(size_t)r1 * DFEAT + ncol] = fmaxf(c1[j] + bias, 0.0f);
    }
}

// ------------------------------------------------- final projection  h @ Wout^T + b
// Block = 16 nodes x 48 padded cols (C=40); 3 waves.
__global__ __launch_bounds__(96) void output_kernel(const float* __restrict__ h,
                                                    const float* __restrict__ Wout,
                                                    const float* __restrict__ bout,
                                                    float* __restrict__ out,
                                                    int n, int cdim) {
    __shared__ float hs[16][LDSPITCH];
    const int base = blockIdx.x * 16;
    const int tid  = threadIdx.x;

    for (int i = tid; i < 16 * DFEAT; i += 96) {
        int row = i >> 7, col = i & (DFEAT - 1);
        int r = base + row;
        int rc = (r < n) ? r : (n - 1);
        hs[row][col] = h[(size_t)rc * DFEAT + col];
    }
    __syncthreads();

    const int wave = tid >> 5;           // 0..2
    const int lane = tid & 31;
    const int half = lane >> 4;
    const int arow = lane & 15;
    const int ncol = wave * 16 + (lane & 15);
    const bool live = (ncol < cdim);

    v8f c = {0.f, 0.f, 0.f, 0.f, 0.f, 0.f, 0.f, 0.f};
    for (int k0 = 0; k0 < DFEAT; k0 += 4) {
        int k = k0 + 2 * half;
        v2f a, b;
        a.x = hs[arow][k];
        a.y = hs[arow][k + 1];
        b.x = live ? Wout[(size_t)ncol * DFEAT + k]     : 0.0f;  // zero-pad N tail
        b.y = live ? Wout[(size_t)ncol * DFEAT + k + 1] : 0.0f;
        c = __builtin_amdgcn_wmma_f32_16x16x4_f32(false, a, false, b, (short)0, c,
                                                  false, false);
    }

    const float bias = live ? bout[ncol] : 0.0f;
    const int m0 = half * 8;
    if (live) {
        for (int j = 0; j < 8; ++j) {
            int r = base + m0 + j;
            if (r < n) out[(size_t)r * cdim + ncol] = c[j] + bias;
        }
    }
}

// ---------------------------------------------------------------- launcher
extern "C" void kernel_launch(void* const* d_in, const int* in_sizes, int n_in,
                              void* d_out, int out_size, void* d_ws, size_t ws_size,
                              hipStream_t stream) {
    const float* x    = (const float*)d_in[0];
    const int*   ei   = (const int*)d_in[1];   // [2][E] row-major
    const float* Wl1  = (const float*)d_in[2];
    const float* bl1  = (const float*)d_in[3];
    const float* Wr1  = (const float*)d_in[4];
    const float* Wl2  = (const float*)d_in[5];
    const float* bl2  = (const float*)d_in[6];
    const float* Wr2  = (const float*)d_in[7];
    const float* Wout = (const float*)d_in[8];
    const float* bout = (const float*)d_in[9];
    float* out = (float*)d_out;

    const int N = in_sizes[0] / DFEAT;
    const int E = in_sizes[1] / 2;
    const int C = in_sizes[9];             // bout length = 40
    const int* src = ei;
    const int* dst = ei + E;

    float* agg = (float*)d_ws;                              // N*128
    float* h1  = agg + (size_t)N * DFEAT;                   // N*128
    float* cnt = h1 + (size_t)N * DFEAT;                    // N

    const long long featElems = (long long)N * DFEAT;
    const int zgridF = 2048;                                // grid-stride zero fill
    const int layerBlocks = (N + MROWS - 1) / MROWS;
    const int headBlocks  = (N + 15) / 16;
    const int scatBlocks  = (E * 32 + 255) / 256;
    const int degBlocks   = (E + 255) / 256;

    // degree counts (shared by both layers)
    zero_kernel<<<256, 256, 0, stream>>>(cnt, N);
    degree_kernel<<<degBlocks, 256, 0, stream>>>(dst, E, cnt);

    // ---- layer 1
    zero_kernel<<<zgridF, 256, 0, stream>>>(agg, featElems);
    scatter_kernel<<<scatBlocks, 256, 0, stream>>>(x, src, dst, agg, E);
    sage_layer_kernel<<<layerBlocks, 256, 0, stream>>>(x, agg, cnt, Wl1, bl1, Wr1, h1, N);

    // ---- layer 2 (output aliases agg: rows staged to LDS before overwrite)
    zero_kernel<<<zgridF, 256, 0, stream>>>(agg, featElems);
    scatter_kernel<<<scatBlocks, 256, 0, stream>>>(h1, src, dst, agg, E);
    sage_layer_kernel<<<layerBlocks, 256, 0, stream>>>(h1, agg, cnt, Wl2, bl2, Wr2, agg, N);

    // ---- classifier head
    output_kernel<<<headBlocks, 96, 0, stream>>>(agg, Wout, bout, out, N, C);
}